// GATNet_26620207301225
// MI455X (gfx1250) — compile-verified
//
#include <hip/hip_runtime.h>
#include <math.h>

typedef float v2f __attribute__((ext_vector_type(2)));
typedef float v8f __attribute__((ext_vector_type(8)));

#define HC 1024   // H * C
#define FD 512    // input feature dim

// ---------------------------------------------------------------------------
// Generic f32 WMMA GEMM:  C[M,Nn] = op(A[M,K]) @ B[K,Nn]  (+bias) (relu)
// One wave computes a 16x64 output strip (4 x 16x16 accumulators) so the
// A fragment is reused by 4 consecutive v_wmma_f32_16x16x4_f32 ops.
// Requires: M % 16 == 0, K % 4 == 0, Nn % 64 == 0.
// ---------------------------------------------------------------------------
template <bool RELU_IN, bool RELU_OUT, bool HAS_BIAS>
__global__ __launch_bounds__(256) void wmma_gemm_f32(
    const float* __restrict__ A, const float* __restrict__ B,
    const float* __restrict__ bias, float* __restrict__ C,
    int M, int K, int Nn) {
  const int lane   = threadIdx.x & 31;
  const int wave   = blockIdx.x * (blockDim.x >> 5) + (threadIdx.x >> 5);
  const int ntiles = Nn >> 6;                 // 64-wide strips
  const int mtile  = wave / ntiles;
  const int ntile  = wave - mtile * ntiles;
  if (mtile * 16 >= M) return;

  const int row  = mtile * 16 + (lane & 15);  // A row this lane feeds
  const int koff = (lane >> 4) * 2;           // lanes 0-15: K=0,1 ; 16-31: K=2,3
  const int col  = ntile * 64 + (lane & 15);  // base B/C column

  v8f acc0 = {}, acc1 = {}, acc2 = {}, acc3 = {};

  const float* __restrict__ arow = A + (size_t)row * K;
  for (int k = 0; k < K; k += 4) {
    v2f a;
    a.x = arow[k + koff];
    a.y = arow[k + koff + 1];
    if (RELU_IN) { a.x = fmaxf(a.x, 0.f); a.y = fmaxf(a.y, 0.f); }

    const float* __restrict__ b0 = B + (size_t)(k + koff) * Nn;
    const float* __restrict__ b1 = b0 + Nn;
    v2f f0, f1, f2, f3;
    f0.x = b0[col];      f0.y = b1[col];
    f1.x = b0[col + 16]; f1.y = b1[col + 16];
    f2.x = b0[col + 32]; f2.y = b1[col + 32];
    f3.x = b0[col + 48]; f3.y = b1[col + 48];

    acc0 = __builtin_amdgcn_wmma_f32_16x16x4_f32(false, a, false, f0, (short)0, acc0, false, false);
    acc1 = __builtin_amdgcn_wmma_f32_16x16x4_f32(false, a, false, f1, (short)0, acc1, false, false);
    acc2 = __builtin_amdgcn_wmma_f32_16x16x4_f32(false, a, false, f2, (short)0, acc2, false, false);
    acc3 = __builtin_amdgcn_wmma_f32_16x16x4_f32(false, a, false, f3, (short)0, acc3, false, false);
  }

  // C/D layout: VGPR r -> row (r) for lanes 0-15, row (r+8) for lanes 16-31
  const int rbase = mtile * 16 + ((lane >> 4) << 3);
  float bb0 = 0.f, bb1 = 0.f, bb2 = 0.f, bb3 = 0.f;
  if (HAS_BIAS) {
    bb0 = bias[col]; bb1 = bias[col + 16]; bb2 = bias[col + 32]; bb3 = bias[col + 48];
  }
#pragma unroll
  for (int r = 0; r < 8; ++r) {
    float v0 = acc0[r] + bb0;
    float v1 = acc1[r] + bb1;
    float v2 = acc2[r] + bb2;
    float v3 = acc3[r] + bb3;
    if (RELU_OUT) {
      v0 = fmaxf(v0, 0.f); v1 = fmaxf(v1, 0.f);
      v2 = fmaxf(v2, 0.f); v3 = fmaxf(v3, 0.f);
    }
    float* cp = C + (size_t)(rbase + r) * Nn + col;
    cp[0] = v0; cp[16] = v1; cp[32] = v2; cp[48] = v3;
  }
}

// ---------------------------------------------------------------------------
// Init: out = bias_conv broadcast per row; m = -inf; denom = 0
// ---------------------------------------------------------------------------
__global__ void init_out_kernel(float* __restrict__ out,
                                const float* __restrict__ bias, int total) {
  int i = blockIdx.x * blockDim.x + threadIdx.x;
  if (i < total) out[i] = bias[i & (HC - 1)];
}

__global__ void init_md_kernel(float* __restrict__ m, float* __restrict__ denom,
                               int total) {
  int i = blockIdx.x * blockDim.x + threadIdx.x;
  if (i < total) { m[i] = -INFINITY; denom[i] = 0.f; }
}

// ---------------------------------------------------------------------------
// Attention coefficients: one warp per (node, head); 512-channel dot product
// ---------------------------------------------------------------------------
__global__ __launch_bounds__(256) void attn_kernel(
    const float* __restrict__ h, const float* __restrict__ att_src,
    const float* __restrict__ att_dst, float* __restrict__ a_src,
    float* __restrict__ a_dst, int n) {
  int warp = (blockIdx.x * blockDim.x + threadIdx.x) >> 5;
  int lane = threadIdx.x & 31;
  int node = warp >> 1;
  int head = warp & 1;
  if (node >= n) return;
  const float* hp = h + (size_t)node * HC + head * 512;
  const float* as = att_src + head * 512;
  const float* ad = att_dst + head * 512;
  float ss = 0.f, sd = 0.f;
  for (int c = lane; c < 512; c += 32) {
    float v = hp[c];
    ss += v * as[c];
    sd += v * ad[c];
  }
#pragma unroll
  for (int off = 16; off; off >>= 1) {
    ss += __shfl_down(ss, off, 32);
    sd += __shfl_down(sd, off, 32);
  }
  if (lane == 0) {
    a_src[node * 2 + head] = ss;
    a_dst[node * 2 + head] = sd;
  }
}

// ---------------------------------------------------------------------------
// Edge pass 1: e = leaky_relu(a_src[s] + a_dst[d]); segment atomic max on dst
// ---------------------------------------------------------------------------
__device__ inline void atomicMaxFloat(float* addr, float val) {
  if (val >= 0.0f)
    atomicMax((int*)addr, __float_as_int(val));
  else
    atomicMin((unsigned int*)addr, __float_as_uint(val));
}

__global__ void edge_logit_max_kernel(const int* __restrict__ ei,
                                      const float* __restrict__ a_src,
                                      const float* __restrict__ a_dst,
                                      float* __restrict__ elog,
                                      float* __restrict__ m, int E, int total) {
  int e = blockIdx.x * blockDim.x + threadIdx.x;
  if (e >= total) return;
  int s, d;
  if (e < E) { s = ei[e]; d = ei[E + e]; } else { s = d = e - E; }
#pragma unroll
  for (int hh = 0; hh < 2; ++hh) {
    float v = a_src[s * 2 + hh] + a_dst[d * 2 + hh];
    v = (v >= 0.f) ? v : 0.2f * v;   // leaky relu, slope 0.2
    elog[e * 2 + hh] = v;
    atomicMaxFloat(&m[d * 2 + hh], v);
  }
}

// ---------------------------------------------------------------------------
// Edge pass 2: ex = exp(e - m[dst]); segment atomic sum into denom
// ---------------------------------------------------------------------------
__global__ void edge_exp_sum_kernel(const int* __restrict__ ei,
                                    float* __restrict__ elog,
                                    const float* __restrict__ m,
                                    float* __restrict__ denom, int E, int total) {
  int e = blockIdx.x * blockDim.x + threadIdx.x;
  if (e >= total) return;
  int d = (e < E) ? ei[E + e] : (e - E);
#pragma unroll
  for (int hh = 0; hh < 2; ++hh) {
    float v = expf(elog[e * 2 + hh] - m[d * 2 + hh]);
    elog[e * 2 + hh] = v;
    atomicAdd(&denom[d * 2 + hh], v);
  }
}

// ---------------------------------------------------------------------------
// Edge pass 3: out[dst] += h[src] * (ex / denom[dst]); one block per edge,
// each of 256 threads handles 4 consecutive channels (float4 gather).
// ---------------------------------------------------------------------------
__global__ __launch_bounds__(256) void aggregate_kernel(
    const int* __restrict__ ei, const float* __restrict__ h,
    const float* __restrict__ elog, const float* __restrict__ denom,
    float* __restrict__ out, int E, int total) {
  int e = blockIdx.x;
  if (e >= total) return;
  int s, d;
  if (e < E) { s = ei[e]; d = ei[E + e]; } else { s = d = e - E; }
  int c = threadIdx.x * 4;            // [c, c+4) within one head (512 % 4 == 0)
  int hh = c >> 9;
  float w = elog[e * 2 + hh] / denom[d * 2 + hh];
  float4 hv = *(const float4*)(h + (size_t)s * HC + c);
  float* op = out + (size_t)d * HC + c;
  atomicAdd(op + 0, hv.x * w);
  atomicAdd(op + 1, hv.y * w);
  atomicAdd(op + 2, hv.z * w);
  atomicAdd(op + 3, hv.w * w);
}

// ---------------------------------------------------------------------------
// Final tiny layer: y = act3[64] @ W3[64,3] + b3 ; pack (y0,y1,y2,||y||^2)
// ---------------------------------------------------------------------------
__global__ void mlp4_kernel(const float* __restrict__ act3,
                            const float* __restrict__ W3,
                            const float* __restrict__ b3,
                            float4* __restrict__ y4, int n) {
  int i = blockIdx.x * blockDim.x + threadIdx.x;
  if (i >= n) return;
  const float* a = act3 + (size_t)i * 64;
  float o0 = b3[0], o1 = b3[1], o2 = b3[2];
#pragma unroll 8
  for (int k = 0; k < 64; ++k) {
    float v = a[k];
    o0 += v * W3[k * 3 + 0];
    o1 += v * W3[k * 3 + 1];
    o2 += v * W3[k * 3 + 2];
  }
  float sq = o0 * o0 + o1 * o1 + o2 * o2;
  y4[i] = make_float4(o0, o1, o2, sq);
}

// ---------------------------------------------------------------------------
// Pairwise distances (gram trick, matching reference):
// d2 = max(sq_i + sq_j - 2*dot, 0); out = d2>0 ? sqrt(d2) : 0
// Block row i = blockIdx.y; each thread writes 4 consecutive j (float4 store).
// ---------------------------------------------------------------------------
__global__ __launch_bounds__(256) void cdist_kernel(
    const float4* __restrict__ y4, float* __restrict__ out, int n) {
  int i = blockIdx.y;
  int j0 = (blockIdx.x * blockDim.x + threadIdx.x) * 4;
  if (j0 >= n) return;
  float4 yi = y4[i];
  float r[4];
#pragma unroll
  for (int u = 0; u < 4; ++u) {
    int j = j0 + u;
    float4 yj = (j < n) ? y4[j] : make_float4(0.f, 0.f, 0.f, 0.f);
    float d2 = yi.w + yj.w -
               2.f * (yi.x * yj.x + yi.y * yj.y + yi.z * yj.z);
    d2 = fmaxf(d2, 0.f);
    r[u] = (d2 > 0.f) ? sqrtf(d2) : 0.f;
  }
  float* op = out + (size_t)i * n + j0;
  if (j0 + 4 <= n) {
    *(float4*)op = make_float4(r[0], r[1], r[2], r[3]);
  } else {
    for (int u = 0; u < 4 && j0 + u < n; ++u) op[u] = r[u];
  }
}

// ---------------------------------------------------------------------------
// Host launcher
// ---------------------------------------------------------------------------
extern "C" void kernel_launch(void* const* d_in, const int* in_sizes, int n_in,
                              void* d_out, int out_size, void* d_ws, size_t ws_size,
                              hipStream_t stream) {
  const float* x         = (const float*)d_in[0];
  const int*   ei        = (const int*)  d_in[1];   // [2,E] flat: row0=src, row1=dst
  const float* W         = (const float*)d_in[2];
  const float* att_src   = (const float*)d_in[3];
  const float* att_dst   = (const float*)d_in[4];
  const float* bias_conv = (const float*)d_in[5];
  const float* Wa        = (const float*)d_in[6];
  const float* ba        = (const float*)d_in[7];
  const float* W1        = (const float*)d_in[8];
  const float* b1        = (const float*)d_in[9];
  const float* W2        = (const float*)d_in[10];
  const float* b2        = (const float*)d_in[11];
  const float* W3        = (const float*)d_in[12];
  const float* b3        = (const float*)d_in[13];

  const int N = in_sizes[0] / FD;   // 10000
  const int E = in_sizes[1] / 2;    // 640000
  const int T = E + N;              // edges + self loops

  // Workspace carve (floats)
  float* ws    = (float*)d_ws;
  float* h     = ws; ws += (size_t)N * HC;   // projected features
  float* outb  = ws; ws += (size_t)N * HC;   // aggregated + bias
  float* a_src = ws; ws += (size_t)N * 2;
  float* a_dst = ws; ws += (size_t)N * 2;
  float* mbuf  = ws; ws += (size_t)N * 2;
  float* denom = ws; ws += (size_t)N * 2;
  float* elog  = ws; ws += (size_t)T * 2;
  float* act1  = ws; ws += (size_t)N * 256;
  float* act2  = ws; ws += (size_t)N * 128;
  float* act3  = ws; ws += (size_t)N * 64;
  float4* y4   = (float4*)ws;               // N float4

  float* dout = (float*)d_out;

  // 1) init out=bias, m=-inf, denom=0
  init_out_kernel<<<(N * HC + 255) / 256, 256, 0, stream>>>(outb, bias_conv, N * HC);
  init_md_kernel<<<(N * 2 + 255) / 256, 256, 0, stream>>>(mbuf, denom, N * 2);

  // 2) h = x @ W   [N,512]@[512,1024]
  {
    int waves = (N / 16) * (HC / 64);
    wmma_gemm_f32<false, false, false><<<(waves + 7) / 8, 256, 0, stream>>>(
        x, W, nullptr, h, N, FD, HC);
  }

  // 3) attention coefficients
  attn_kernel<<<(N * 2 * 32 + 255) / 256, 256, 0, stream>>>(h, att_src, att_dst,
                                                            a_src, a_dst, N);

  // 4) segment softmax over destinations
  edge_logit_max_kernel<<<(T + 255) / 256, 256, 0, stream>>>(ei, a_src, a_dst,
                                                             elog, mbuf, E, T);
  edge_exp_sum_kernel<<<(T + 255) / 256, 256, 0, stream>>>(ei, elog, mbuf, denom,
                                                           E, T);

  // 5) weighted scatter aggregation
  aggregate_kernel<<<T, 256, 0, stream>>>(ei, h, elog, denom, outb, E, T);

  // 6) MLP (ReLU fused)
  {
    int waves = (N / 16) * (256 / 64);
    wmma_gemm_f32<true, true, true><<<(waves + 7) / 8, 256, 0, stream>>>(
        outb, Wa, ba, act1, N, HC, 256);
  }
  {
    int waves = (N / 16) * (128 / 64);
    wmma_gemm_f32<false, true, true><<<(waves + 7) / 8, 256, 0, stream>>>(
        act1, W1, b1, act2, N, 256, 128);
  }
  {
    int waves = (N / 16) * (64 / 64);
    wmma_gemm_f32<false, true, true><<<(waves + 7) / 8, 256, 0, stream>>>(
        act2, W2, b2, act3, N, 128, 64);
  }
  mlp4_kernel<<<(N + 255) / 256, 256, 0, stream>>>(act3, W3, b3, y4, N);

  // 7) pairwise distance matrix [N,N]
  {
    int j4 = (N + 3) / 4;
    dim3 grid((j4 + 255) / 256, N);
    cdist_kernel<<<grid, 256, 0, stream>>>(y4, dout, N);
  }
}